// RotaryPositionalEmbeddings2D_14508399526741
// MI455X (gfx1250) — compile-verified
//
#include <hip/hip_runtime.h>
#include <cstdint>

// Problem constants (match reference setup_inputs()).
#define NB    32           // batch
#define NS    1024         // sequence
#define NH    16           // heads
#define ND    64           // head dim
#define MAXG  64           // max grid size (position cache extent)
#define CHUNK 16           // s-rows per block
#define NBUF  4            // async pipeline depth (rows in flight per wave)
#define ROW_FLOATS   (NH * ND)    // 1024 floats = 4KB per (b,s) row
#define TABLE_FLOATS (MAXG * ND)  // 64 pos * 32 pairs * {cos,sin} = 4096 floats = 16KB
#define LOG2_BASE 13.287712379549449f   // log2(10000)

// ---------------------------------------------------------------------------
// Kernel 1: build the cos/sin table once into global scratch (stays L2-hot).
// Layout: table[pos*64 + k*2 + {0:cos, 1:sin}], pos in [0,64), k in [0,32).
// ---------------------------------------------------------------------------
__global__ __launch_bounds__(256) void rope2d_build_table(float* __restrict__ table) {
    const int tid = threadIdx.x;
#pragma unroll
    for (int n = 0; n < 8; ++n) {
        const int idx = tid + n * 256;       // 0..2047
        const int pos = idx >> 5;            // 0..63
        const int k   = idx & 31;            // 0..31
        const float theta = exp2f(-(float)k * (LOG2_BASE / 32.0f));
        float sv, cv;
        sincosf((float)pos * theta, &sv, &cv);
        table[(pos << 6) + (k << 1) + 0] = cv;
        table[(pos << 6) + (k << 1) + 1] = sv;
    }
}

// ---------------------------------------------------------------------------
// Kernel 2: streaming 2D-RoPE. All inbound data (table + x rows) moves via
// the gfx1250 async global->LDS path; rotation math reads LDS, results go
// out as coalesced b128 stores. Depth-NBUF pipeline per wave.
// ---------------------------------------------------------------------------
template <bool USE_TABLE>
__global__ __launch_bounds__(256) void rope2d_main(
    const float* __restrict__ x,
    const int*   __restrict__ grid_sizes,
    const float* __restrict__ table,
    float*       __restrict__ out)
{
    __shared__ __align__(16) float rope[TABLE_FLOATS];       // 16KB
    __shared__ __align__(16) float xbuf[NBUF][ROW_FLOATS];   // 16KB

    const int tid = threadIdx.x;            // 0..255
    const int b   = blockIdx.y;             // 0..31
    const int s0  = blockIdx.x * CHUNK;
    const unsigned cols = (unsigned)grid_sizes[2 * b + 1];

    const size_t rowbase = ((size_t)b * NS + (size_t)s0) * ROW_FLOATS;
    const float* gsrc = x + rowbase + (size_t)tid * 4;

    // Wave-relative LDS byte offsets of this thread's 16B slots.
    const uint32_t xoff = (uint32_t)(uintptr_t)(void*)&xbuf[0][tid * 4];
    const uint32_t roff = (uint32_t)(uintptr_t)(void*)&rope[tid * 4];

    if constexpr (USE_TABLE) {
        // Stage the 16KB table via 4 async b128 chunks per thread (L2-hot source).
        const unsigned long long ta = (unsigned long long)(uintptr_t)(table + (size_t)tid * 4);
#pragma unroll
        for (int n = 0; n < 4; ++n) {
            const uint32_t lo = roff + (uint32_t)(n * 4096);
            const unsigned long long ga = ta + (unsigned long long)(n * 4096);
            asm volatile("global_load_async_to_lds_b128 %0, %1, off"
                         :: "v"(lo), "v"(ga) : "memory");
        }
    }

    // Prologue: issue async loads for x rows 0..NBUF-1.
#pragma unroll
    for (int r = 0; r < NBUF; ++r) {
        const uint32_t lo = xoff + (uint32_t)(r * ROW_FLOATS * sizeof(float));
        const unsigned long long ga =
            (unsigned long long)(uintptr_t)(gsrc + (size_t)r * ROW_FLOATS);
        asm volatile("global_load_async_to_lds_b128 %0, %1, off"
                     :: "v"(lo), "v"(ga) : "memory");
    }

    if constexpr (USE_TABLE) {
        // Async completions are in-order per wave: allowing NBUF outstanding
        // guarantees the 4 table chunks (issued first) have landed.
        asm volatile("s_wait_asynccnt 4" ::: "memory");
    } else {
        // Fallback: build the table in-LDS (overlaps with the x-row DMAs).
#pragma unroll
        for (int n = 0; n < 8; ++n) {
            const int idx = tid + n * 256;
            const int pos = idx >> 5;
            const int k   = idx & 31;
            const float theta = exp2f(-(float)k * (LOG2_BASE / 32.0f));
            float sv, cv;
            sincosf((float)pos * theta, &sv, &cv);
            rope[(pos << 6) + (k << 1) + 0] = cv;
            rope[(pos << 6) + (k << 1) + 1] = sv;
        }
    }
    __syncthreads();   // table visible to all waves

    // Incremental (row,col) position: one division per block, then carry.
    unsigned i = (unsigned)s0 / cols;
    unsigned j = (unsigned)s0 - i * cols;
    const int q = tid & 15;   // pair-quad index within head

#pragma unroll
    for (int r = 0; r < CHUNK; ++r) {
        // Retire the oldest outstanding x-row (constants fold under unroll).
        const int pending = (CHUNK - r) < NBUF ? (CHUNK - r) : NBUF;
        if      (pending >= 4) asm volatile("s_wait_asynccnt 3" ::: "memory");
        else if (pending == 3) asm volatile("s_wait_asynccnt 2" ::: "memory");
        else if (pending == 2) asm volatile("s_wait_asynccnt 1" ::: "memory");
        else                   asm volatile("s_wait_asynccnt 0" ::: "memory");

        // This wave reads exactly the region its own async op wrote -> no barrier.
        const float4 v  = *(const float4*)&xbuf[r % NBUF][tid * 4];
        const float4 rr = *(const float4*)&rope[(i << 6) + (q << 2)];  // pos i: cos/sin @ 2q,2q+1
        const float4 rc = *(const float4*)&rope[(j << 6) + (q << 2)];  // pos j

        // pair 2q: row rotation then col rotation
        float y0 = v.x * rr.x - v.y * rr.y;
        float y1 = v.y * rr.x + v.x * rr.y;
        float z0 = y0 * rc.x - y1 * rc.y;
        float z1 = y1 * rc.x + y0 * rc.y;
        // pair 2q+1
        float w0 = v.z * rr.z - v.w * rr.w;
        float w1 = v.w * rr.z + v.z * rr.w;
        float u0 = w0 * rc.z - w1 * rc.w;
        float u1 = w1 * rc.z + w0 * rc.w;

        float4 o; o.x = z0; o.y = z1; o.z = u0; o.w = u1;
        *(float4*)(out + rowbase + (size_t)r * ROW_FLOATS + (size_t)tid * 4) = o;

        // Refill the just-consumed buffer with row r+NBUF.
        if (r + NBUF < CHUNK) {
            // Async LDS writes are unordered vs this wave's ds reads: drain DScnt
            // so the b128 reads above have fully left LDS before the overwrite.
            asm volatile("s_wait_dscnt 0" ::: "memory");
            const uint32_t lo = xoff + (uint32_t)((r % NBUF) * ROW_FLOATS * sizeof(float));
            const unsigned long long ga =
                (unsigned long long)(uintptr_t)(gsrc + (size_t)(r + NBUF) * ROW_FLOATS);
            asm volatile("global_load_async_to_lds_b128 %0, %1, off"
                         :: "v"(lo), "v"(ga) : "memory");
        }

        // Advance flat position -> (i,j) with a carry (no per-iteration division).
        j += 1;
        if (j == cols) { j = 0; i += 1; }
    }
}

extern "C" void kernel_launch(void* const* d_in, const int* in_sizes, int n_in,
                              void* d_out, int out_size, void* d_ws, size_t ws_size,
                              hipStream_t stream) {
    const float* x   = (const float*)d_in[0];   // [32,1024,16,64] f32
    const int*   gs  = (const int*)d_in[1];     // [32,2] i32
    float*       out = (float*)d_out;           // [32,1024,16,64] f32
    (void)in_sizes; (void)n_in; (void)out_size;

    dim3 grid(NS / CHUNK, NB);   // (64, 32) = 2048 blocks
    dim3 block(256);             // 8 waves; one 4KB (b,s) row per iteration

    if (ws_size >= (size_t)TABLE_FLOATS * sizeof(float)) {
        float* table = (float*)d_ws;
        rope2d_build_table<<<dim3(1), dim3(256), 0, stream>>>(table);
        rope2d_main<true><<<grid, block, 0, stream>>>(x, gs, table, out);
    } else {
        rope2d_main<false><<<grid, block, 0, stream>>>(x, gs, nullptr, out);
    }
}